// Focal_loss_73151882985870
// MI455X (gfx1250) — compile-verified
//
#include <hip/hip_runtime.h>
#include <hip/hip_bf16.h>
#include <stdint.h>

// ---------------- problem constants (match reference) ----------------
#define BB      4
#define TT      512
#define VOCAB   32000
#define NSPLIT  50                 // vocab splits per batch row
#define VPER    (VOCAB / NSPLIT)   // 640 vocab rows per block
#define CHUNK   8                  // rows per async chunk (8*512*4 = 16 KB)
#define NCHUNK  (VPER / CHUNK)     // 80 chunks
#define IGNORE_WRAP (VOCAB - 100)  // jnp .at[-100] wraps to V-100

#define LS      0.1f               // label smoothing
#define MARGINF 0.6f
#define SCALEF  0.1f
#define KKF     1000000.0f

// ---------------- gfx1250 async global->LDS path ----------------
#if defined(__AMDGCN__) && __has_builtin(__builtin_amdgcn_global_load_async_to_lds_b128)
#define HAVE_ASYNC 1
#else
#define HAVE_ASYNC 0
#endif

#if HAVE_ASYNC
#if __has_builtin(__builtin_amdgcn_s_wait_asynccnt)
#define WAIT_ASYNC(n) __builtin_amdgcn_s_wait_asynccnt(n)
#else
#define WAIT_ASYNC(n) asm volatile("s_wait_asynccnt %0" ::"i"(n))
#endif

typedef int v4i __attribute__((ext_vector_type(4)));
typedef __attribute__((address_space(1))) v4i ga_v4i;   // global (AS1) int4
typedef __attribute__((address_space(3))) v4i la_v4i;   // LDS (AS3) int4

// Copy one contiguous 16 KB chunk (CHUNK*TT floats) global -> LDS.
// Each of 512 threads moves 32 bytes = 2x b128 ops (asynccnt += 2 per wave).
__device__ __forceinline__ void issue_chunk(const float* g, float* l, int t) {
  uintptr_t ga = (uintptr_t)g + (uintptr_t)(t * 16);
  unsigned  la = (unsigned)(uintptr_t)l + (unsigned)(t * 16);
  __builtin_amdgcn_global_load_async_to_lds_b128((ga_v4i*)ga, (la_v4i*)la, 0, 0);
  __builtin_amdgcn_global_load_async_to_lds_b128((ga_v4i*)(ga + 8192),
                                                 (la_v4i*)(la + 8192u), 0, 0);
}
#endif

// ---------------- kernel 1: streaming online-softmax partials ----------------
// grid = BB*NSPLIT blocks of TT threads. Block (b, sp) scans vocab rows
// [sp*VPER, (sp+1)*VPER) for all t columns. For fixed b the (v,t) plane is
// contiguous, so every CHUNK rows form one contiguous 16 KB slab.
__global__ __launch_bounds__(TT) void k1(const float* __restrict__ inp,
                                         float* __restrict__ pm,
                                         float* __restrict__ ps,
                                         float* __restrict__ psx,
                                         int*   __restrict__ pam) {
  const int t  = threadIdx.x;
  const int b  = blockIdx.x / NSPLIT;
  const int sp = blockIdx.x % NSPLIT;
  const int v0 = sp * VPER;
  const float* base = inp + ((size_t)b * VOCAB + v0) * TT;

  float m = -INFINITY, s = 0.0f, sx = 0.0f;
  int am = v0;

#if HAVE_ASYNC
  __shared__ float lds[2][CHUNK * TT];   // 2 x 16 KB double buffer

  issue_chunk(base, &lds[0][0], t);
  for (int c = 0; c < NCHUNK; ++c) {
    if (c + 1 < NCHUNK)
      issue_chunk(base + (size_t)(c + 1) * CHUNK * TT, &lds[(c + 1) & 1][0], t);
    // async loads retire in order: oldest 2 = chunk c
    if (c + 1 < NCHUNK) WAIT_ASYNC(2); else WAIT_ASYNC(0);
    __syncthreads();                     // all waves' chunk-c data visible
    const float* buf = &lds[c & 1][0];
    const int vbase = v0 + c * CHUNK;
#pragma unroll
    for (int vv = 0; vv < CHUNK; ++vv) {
      float x = buf[vv * TT + t];
      sx += x;
      if (x > m) { s = s * expf(m - x) + 1.0f; m = x; am = vbase + vv; }
      else       { s += expf(x - m); }
    }
    __syncthreads();                     // done reading before buffer reuse
  }
#else
  for (int v = 0; v < VPER; ++v) {
    __builtin_prefetch(base + (size_t)(v + 8) * TT + t, 0, 0);
    float x = base[(size_t)v * TT + t];
    sx += x;
    if (x > m) { s = s * expf(m - x) + 1.0f; m = x; am = v0 + v; }
    else       { s += expf(x - m); }
  }
#endif

  const int idx = (b * NSPLIT + sp) * TT + t;
  pm[idx] = m; ps[idx] = s; psx[idx] = sx; pam[idx] = am;
}

// ---------------- kernel 2: combine + loss + counts + sentence scale ----------
__global__ __launch_bounds__(TT) void k2(const float* __restrict__ inp,
                                         const float* __restrict__ w,
                                         const int* __restrict__ tgt,
                                         const int* __restrict__ slen,
                                         const float* __restrict__ pm,
                                         const float* __restrict__ ps,
                                         const float* __restrict__ psx,
                                         const int* __restrict__ pam,
                                         float* __restrict__ loss,
                                         int* __restrict__ tcnt,
                                         int* __restrict__ acnt) {
  const int t = threadIdx.x, b = blockIdx.x;

  float m = -INFINITY, s = 0.0f, sx = 0.0f;
  int am = 0;
  for (int sp = 0; sp < NSPLIT; ++sp) {       // ascending v -> first-occurrence argmax
    const int idx = (b * NSPLIT + sp) * TT + t;
    const float mi = pm[idx], si = ps[idx];
    sx += psx[idx];
    if (mi > m) { s = s * expf(m - mi) + si; am = pam[idx]; m = mi; }
    else        { s += si * expf(mi - m); }
  }
  const float lse = m + logf(s);

  const int tg = tgt[b * TT + t];             // targets are in [0, V)
  const float xt = inp[((size_t)b * VOCAB + tg) * TT + t];
  const float nll = lse - xt;
  const float occ = expf(xt - lse);
  const float sum_logp = sx - (float)VOCAB * lse;
  const float fmean = (-sum_logp - nll) * (1.0f / (float)(VOCAB - 1));
  float lp = (nll * (1.0f - LS) + fmean * LS) * w[tg];
  const float om = 1.0f - occ;                // gamma = 2
  lp *= om * om;

  atomicAdd(&acnt[tg], 1);
  if (am == tg) atomicAdd(&tcnt[tg], 1);

  // sentence weight: max over valid t of sigmoid(K*(MARGIN - occ))
  const float sw = (t < slen[b]) ? 1.0f / (1.0f + expf(KKF * (occ - MARGINF))) : 0.0f;
  __shared__ float red[TT];
  red[t] = sw;
  __syncthreads();
  for (int o = TT / 2; o > 0; o >>= 1) {
    if (t < o) red[t] = fmaxf(red[t], red[t + o]);
    __syncthreads();
  }
  const float sws = fminf(fmaxf(red[0], SCALEF), 1.0f);
  loss[b * TT + t] = lp * sws;
}

// ---------------- helpers ----------------
__global__ void k_zero(int* __restrict__ p, int n) {
  int i = blockIdx.x * blockDim.x + threadIdx.x;
  if (i < n) p[i] = 0;
}
__global__ void k_fix(int* __restrict__ tcnt, int* __restrict__ acnt) {
  tcnt[IGNORE_WRAP] = 0;
  acnt[IGNORE_WRAP] = 0;
}

extern "C" void kernel_launch(void* const* d_in, const int* in_sizes, int n_in,
                              void* d_out, int out_size, void* d_ws, size_t ws_size,
                              hipStream_t stream) {
  const float* inp  = (const float*)d_in[0];   // (B, V, T) f32
  const float* w    = (const float*)d_in[1];   // (V,) f32
  const int*   tgt  = (const int*)d_in[2];     // (B, T) int
  const int*   slen = (const int*)d_in[3];     // (B,) int

  float* loss = (float*)d_out;                 // B*T floats
  int*   tcnt = (int*)d_out + BB * TT;         // V int32
  int*   acnt = tcnt + VOCAB;                  // V int32

  const size_t NP = (size_t)BB * NSPLIT * TT;  // 102400 partials per array
  float* pm  = (float*)d_ws;
  float* ps  = pm + NP;
  float* psx = ps + NP;
  int*   pam = (int*)(psx + NP);               // total 1.6 MB of d_ws

  k_zero<<<(2 * VOCAB + 255) / 256, 256, 0, stream>>>(tcnt, 2 * VOCAB);
  k1<<<BB * NSPLIT, TT, 0, stream>>>(inp, pm, ps, psx, pam);
  k2<<<BB, TT, 0, stream>>>(inp, w, tgt, slen, pm, ps, psx, pam, loss, tcnt, acnt);
  k_fix<<<1, 1, 0, stream>>>(tcnt, acnt);
}